// TopKGate_13623636263395
// MI455X (gfx1250) — compile-verified
//
#include <hip/hip_runtime.h>
#include <hip/hip_bf16.h>
#include <float.h>

// MoE top-2 gate (Tutel-style), expert_fn = identity, world_size = 1.
// The dispatch/gather round-trip collapses to out[s,:] = scale[s] * x[s,:]
// with scale[s] = sum_k gates_n[s,k] * valid[s,k].   Plus scalar l_loss.
//
// d_in[0] = x  [8192, 4096] f32
// d_in[1] = wg [  64, 4096] f32
// d_out   = out [8192*4096] f32  ++  l_loss [1] f32
//
// Roofline: dominant traffic = x read twice + out written once ~= 384 MB
// @ 23.3 TB/s ~= 16.5 us. GEMM is 4.3 GFLOP fp32 -> V_WMMA_F32_16X16X4_F32,
// staged via async global->LDS double buffering (ASYNCcnt).

typedef __attribute__((ext_vector_type(2))) float v2f;
typedef __attribute__((ext_vector_type(8))) float v8f;

#define S_TOKENS 8192
#define MDIM     4096
#define NEXP     64
#define CAPACITY 256   // TOP_K * int(CAP_FACTOR * ceil(S/E)) = 2*128
#define KC       64    // K-chunk per LDS stage
#define NCHUNK   (MDIM / KC)

// Async global->LDS copy of 16 bytes (GLOBAL_LOAD_ASYNC_TO_LDS_B128).
// LDS address = low 32 bits of the generic pointer (flat aperture rule:
// address truncated to 32 bits); global address = 64-bit VGPR pair, saddr=off.
__device__ __forceinline__ void async_b128(const float* g, float* l) {
  asm volatile("global_load_async_to_lds_b128 %0, %1, off"
               :: "v"((unsigned)(unsigned long long)l),
                  "v"((unsigned long long)g)
               : "memory");
}

// ---------------------------------------------------------------------------
// Kernel 1: logits[s,e] = sum_m x[s,m] * wg[e,m]   via V_WMMA_F32_16X16X4_F32
// Block: 128 threads (4 waves). Block tile: 16 tokens x 64 experts.
// Wave w computes the 16x16 tile for experts [w*16, w*16+16).
// Double-buffered async staging; two WMMA accumulator chains.
// ---------------------------------------------------------------------------
__global__ __launch_bounds__(128)
void k_gate_gemm(const float* __restrict__ x, const float* __restrict__ wg,
                 float* __restrict__ logits) {
  __shared__ float As[2][16][68];   // x tile  [16 tokens x 64 k] (+pad -> stride 272B)
  __shared__ float Bs[2][64][68];   // wg tile [64 experts x 64 k]

  const int tid  = threadIdx.x;
  const int wave = tid >> 5;     // 0..3  -> expert tile
  const int lane = tid & 31;
  const int s0   = blockIdx.x * 16;

  const int a_row = tid >> 3;          // 0..15, 8 threads per row (2 x b128 each)
  const int a_col = (tid & 7) * 8;
  const int b_row = tid >> 1;          // 0..63, 2 threads per row (8 x b128 each)
  const int b_col = (tid & 1) * 32;

  const float* xrow = x  + (size_t)(s0 + a_row) * MDIM + a_col;
  const float* wrow = wg + (size_t)b_row      * MDIM + b_col;

  // 10 async b128 copies per thread per chunk (2 for x, 8 for wg).
  auto issue = [&](int buf, int m0) {
    async_b128(xrow + m0,     &As[buf][a_row][a_col]);
    async_b128(xrow + m0 + 4, &As[buf][a_row][a_col + 4]);
    #pragma unroll
    for (int i = 0; i < 8; ++i)
      async_b128(wrow + m0 + i * 4, &Bs[buf][b_row][b_col + i * 4]);
  };

  // Fragment geometry (ISA 7.12.2):
  //  A (16x4, MxK): lane<16 holds row=lane, {K0,K1}; lane>=16 -> row=lane-16, {K2,K3}
  //  B (4x16, KxN): lane<16 holds col=lane, {K0,K1}; lane>=16 -> col=lane-16, {K2,K3}
  const int frag_r = lane & 15;
  const int koff   = (lane >> 4) << 1;  // 0 or 2

  v8f acc0 = {};
  v8f acc1 = {};

  issue(0, 0);
  for (int c = 0; c < NCHUNK; ++c) {
    const int buf = c & 1;
    if (c + 1 < NCHUNK) {
      issue(buf ^ 1, (c + 1) * KC);
      // Async loads retire in order: waiting <=10 guarantees the 10 copies
      // of chunk c (this buffer) have landed in LDS.
      asm volatile("s_wait_asynccnt 10" ::: "memory");
    } else {
      asm volatile("s_wait_asynccnt 0" ::: "memory");
    }
    __syncthreads();

    #pragma unroll
    for (int kk = 0; kk < 16; kk += 2) {
      const int k0 = kk * 4 + koff;
      const int k1 = k0 + 4;
      v2f a0, b0, a1, b1;
      a0.x = As[buf][frag_r][k0];
      a0.y = As[buf][frag_r][k0 + 1];
      b0.x = Bs[buf][wave * 16 + frag_r][k0];
      b0.y = Bs[buf][wave * 16 + frag_r][k0 + 1];
      a1.x = As[buf][frag_r][k1];
      a1.y = As[buf][frag_r][k1 + 1];
      b1.x = Bs[buf][wave * 16 + frag_r][k1];
      b1.y = Bs[buf][wave * 16 + frag_r][k1 + 1];
      // (neg_a, A, neg_b, B, c_mod, C, reuse_a, reuse_b)
      acc0 = __builtin_amdgcn_wmma_f32_16x16x4_f32(false, a0, false, b0,
                                                   (short)0, acc0, false, false);
      acc1 = __builtin_amdgcn_wmma_f32_16x16x4_f32(false, a1, false, b1,
                                                   (short)0, acc1, false, false);
    }
    __syncthreads();
  }

  // C/D layout: element (M, N) with M = vgpr + 8*(lane>=16), N = lane&15.
  const int n  = lane & 15;
  const int mo = (lane >> 4) * 8;
  #pragma unroll
  for (int r = 0; r < 8; ++r)
    logits[(size_t)(s0 + mo + r) * NEXP + wave * 16 + n] = acc0[r] + acc1[r];
}

// ---------------------------------------------------------------------------
// Kernel 2: per-token softmax over 64 experts + top-2 + normalized gate pair.
// One wave32 per token (lane handles experts {lane, lane+32}); 8 tokens/block.
// ---------------------------------------------------------------------------
__global__ __launch_bounds__(256)
void k_softmax_top2(const float* __restrict__ logits,
                    float* __restrict__ gates,
                    int* __restrict__ idx0, int* __restrict__ idx1,
                    float* __restrict__ w0, float* __restrict__ w1) {
  const int wave = threadIdx.x >> 5;
  const int lane = threadIdx.x & 31;
  const int t    = blockIdx.x * 8 + wave;

  const float* row = logits + (size_t)t * NEXP;
  float v0 = row[lane];
  float v1 = row[lane + 32];

  // softmax
  float mx = fmaxf(v0, v1);
  #pragma unroll
  for (int off = 16; off > 0; off >>= 1)
    mx = fmaxf(mx, __shfl_xor(mx, off, 32));
  float e0 = __expf(v0 - mx);
  float e1 = __expf(v1 - mx);
  float sm = e0 + e1;
  #pragma unroll
  for (int off = 16; off > 0; off >>= 1)
    sm += __shfl_xor(sm, off, 32);
  const float inv = 1.0f / sm;
  const float g0 = e0 * inv, g1 = e1 * inv;
  gates[(size_t)t * NEXP + lane]      = g0;
  gates[(size_t)t * NEXP + 32 + lane] = g1;

  // top-1 (ties -> lower index, matching lax.top_k)
  float bv; int bi;
  if (v0 >= v1) { bv = v0; bi = lane; } else { bv = v1; bi = lane + 32; }
  #pragma unroll
  for (int off = 16; off > 0; off >>= 1) {
    float ov = __shfl_xor(bv, off, 32);
    int   oi = __shfl_xor(bi, off, 32);
    if (ov > bv || (ov == bv && oi < bi)) { bv = ov; bi = oi; }
  }
  const int i0 = bi;

  // top-2: exclude i0
  float m0v = (i0 == lane)      ? -FLT_MAX : v0;
  float m1v = (i0 == lane + 32) ? -FLT_MAX : v1;
  if (m0v >= m1v) { bv = m0v; bi = lane; } else { bv = m1v; bi = lane + 32; }
  #pragma unroll
  for (int off = 16; off > 0; off >>= 1) {
    float ov = __shfl_xor(bv, off, 32);
    int   oi = __shfl_xor(bi, off, 32);
    if (ov > bv || (ov == bv && oi < bi)) { bv = ov; bi = oi; }
  }
  const int i1 = bi;

  // fetch softmax values at i0,i1 (i0,i1 uniform across the wave)
  const float gi0 = (i0 < 32) ? __shfl(g0, i0, 32) : __shfl(g1, i0 - 32, 32);
  const float gi1 = (i1 < 32) ? __shfl(g0, i1, 32) : __shfl(g1, i1 - 32, 32);
  const float den = fmaxf(gi0 + gi1, FLT_EPSILON);   // postnorm (top_k > 1)

  if (lane == 0) {
    idx0[t] = i0;  idx1[t] = i1;
    w0[t] = gi0 / den;  w1[t] = gi1 / den;
  }
}

// ---------------------------------------------------------------------------
// Kernel 3: me[e] = sum_s gates[s,e]  (fixed-order tree -> deterministic)
// ---------------------------------------------------------------------------
__global__ __launch_bounds__(256)
void k_me_reduce(const float* __restrict__ gates, float* __restrict__ me) {
  __shared__ float red[256];
  const int e = blockIdx.x;
  float acc = 0.0f;
  for (int s = threadIdx.x; s < S_TOKENS; s += 256)
    acc += gates[(size_t)s * NEXP + e];
  red[threadIdx.x] = acc;
  __syncthreads();
  #pragma unroll
  for (int off = 128; off > 0; off >>= 1) {
    if (threadIdx.x < off) red[threadIdx.x] += red[threadIdx.x + off];
    __syncthreads();
  }
  if (threadIdx.x == 0) me[e] = red[0];
}

// ---------------------------------------------------------------------------
// Kernel 4: capacity scan. Inherently sequential in token order per
// (k, expert) bin: 64 bins per phase, one thread each, idx staged in LDS
// (lockstep same-address reads -> bank broadcast). Also emits ce + l_loss.
// ---------------------------------------------------------------------------
__global__ __launch_bounds__(128)
void k_capacity_scan(const int* __restrict__ idx0, const int* __restrict__ idx1,
                     const float* __restrict__ me,
                     int* __restrict__ v0flag, int* __restrict__ v1flag,
                     float* __restrict__ loss_out) {
  __shared__ int   sidx[S_TOKENS];  // 32 KB
  __shared__ int   cnt0[NEXP];
  __shared__ float red[NEXP];
  const int tid = threadIdx.x;

  for (int s = tid; s < S_TOKENS; s += 128) sidx[s] = idx0[s];
  __syncthreads();
  if (tid < NEXP) {                 // k = 0 bins
    int cnt = 0;
    for (int s = 0; s < S_TOKENS; ++s)
      if (sidx[s] == tid) { v0flag[s] = (cnt < CAPACITY) ? 1 : 0; ++cnt; }
    cnt0[tid] = cnt;                // ce[e] (pre-clip count of top-1 picks)
  }
  __syncthreads();
  for (int s = tid; s < S_TOKENS; s += 128) sidx[s] = idx1[s];
  __syncthreads();
  if (tid < NEXP) {                 // k = 1 bins, offset by counts[0][e]
    int cnt = cnt0[tid];
    for (int s = 0; s < S_TOKENS; ++s)
      if (sidx[s] == tid) { v1flag[s] = (cnt < CAPACITY) ? 1 : 0; ++cnt; }
  }
  __syncthreads();
  if (tid < NEXP) red[tid] = me[tid] * (float)cnt0[tid];
  __syncthreads();
  if (tid == 0) {
    float sum = 0.0f;
    for (int e = 0; e < NEXP; ++e) sum += red[e];
    *loss_out = sum * ((float)NEXP / ((float)S_TOKENS * (float)S_TOKENS));
  }
}

// ---------------------------------------------------------------------------
// Kernel 5: out[s,:] = (w0*valid0 + w1*valid1) * x[s,:]
// ---------------------------------------------------------------------------
__global__ __launch_bounds__(256)
void k_combine(const float* __restrict__ x,
               const float* __restrict__ w0, const float* __restrict__ w1,
               const int* __restrict__ v0flag, const int* __restrict__ v1flag,
               float* __restrict__ out) {
  __shared__ float ssc;
  const int s = blockIdx.x;
  if (threadIdx.x == 0)
    ssc = w0[s] * (float)v0flag[s] + w1[s] * (float)v1flag[s];
  __syncthreads();
  const float sc = ssc;
  const float4* xi = (const float4*)(x + (size_t)s * MDIM);
  float4*       xo = (float4*)(out + (size_t)s * MDIM);
  for (int j = threadIdx.x; j < MDIM / 4; j += 256) {
    float4 v = xi[j];
    v.x *= sc; v.y *= sc; v.z *= sc; v.w *= sc;
    xo[j] = v;
  }
}

// ---------------------------------------------------------------------------
extern "C" void kernel_launch(void* const* d_in, const int* in_sizes, int n_in,
                              void* d_out, int out_size, void* d_ws, size_t ws_size,
                              hipStream_t stream) {
  (void)in_sizes; (void)n_in; (void)out_size; (void)ws_size;
  const float* x  = (const float*)d_in[0];
  const float* wg = (const float*)d_in[1];
  float* out      = (float*)d_out;
  float* loss_out = out + (size_t)S_TOKENS * MDIM;   // tuple output #2

  char* ws = (char*)d_ws;
  size_t off = 0;
  float* logits = (float*)(ws + off); off += (size_t)S_TOKENS * NEXP * sizeof(float);
  float* gates  = (float*)(ws + off); off += (size_t)S_TOKENS * NEXP * sizeof(float);
  int*   idx0   = (int*)  (ws + off); off += (size_t)S_TOKENS * sizeof(int);
  int*   idx1   = (int*)  (ws + off); off += (size_t)S_TOKENS * sizeof(int);
  float* w0     = (float*)(ws + off); off += (size_t)S_TOKENS * sizeof(float);
  float* w1     = (float*)(ws + off); off += (size_t)S_TOKENS * sizeof(float);
  int*   v0f    = (int*)  (ws + off); off += (size_t)S_TOKENS * sizeof(int);
  int*   v1f    = (int*)  (ws + off); off += (size_t)S_TOKENS * sizeof(int);
  float* me     = (float*)(ws + off); off += (size_t)NEXP * sizeof(float);

  k_gate_gemm    <<<S_TOKENS / 16, 128, 0, stream>>>(x, wg, logits);
  k_softmax_top2 <<<S_TOKENS / 8, 256, 0, stream>>>(logits, gates, idx0, idx1, w0, w1);
  k_me_reduce    <<<NEXP, 256, 0, stream>>>(gates, me);
  k_capacity_scan<<<1, 128, 0, stream>>>(idx0, idx1, me, v0f, v1f, loss_out);
  k_combine      <<<S_TOKENS, 256, 0, stream>>>(x, w0, w1, v0f, v1f, out);
}